// GRUWrapper_30150670418562
// MI455X (gfx1250) — compile-verified
//
#include <hip/hip_runtime.h>

typedef __attribute__((ext_vector_type(2))) float v2f;
typedef __attribute__((ext_vector_type(8))) float v8f;

#define T_DIM 256
#define B_DIM 128
#define E_DIM 512
#define H_DIM 512
#define G_DIM 1536   // 3*H
#define LN_EPS 1e-5f

__device__ __forceinline__ float warp_allreduce_sum(float v) {
#pragma unroll
  for (int off = 16; off > 0; off >>= 1) v += __shfl_xor(v, off, 32);
  return v;
}

__device__ __forceinline__ v8f wmma_f32(v2f a, v2f b, v8f c) {
  return __builtin_amdgcn_wmma_f32_16x16x4_f32(false, a, false, b, (short)0, c,
                                               false, false);
}

// ---------------------------------------------------------------------------
// Kernel 1: per-row mean / rstd of x viewed as (T*B, E). One wave per row.
// ---------------------------------------------------------------------------
__global__ __launch_bounds__(256) void rowstats_kernel(
    const float* __restrict__ x, float2* __restrict__ stats) {
  const int wid = threadIdx.x >> 5;
  const int lane = threadIdx.x & 31;
  const int row = blockIdx.x * 8 + wid;
  const float* xr = x + (size_t)row * E_DIM;
  float s = 0.f, ss = 0.f;
#pragma unroll
  for (int j = 0; j < 4; ++j) {
    float4 v = *(const float4*)(xr + lane * 4 + j * 128);
    s += v.x + v.y + v.z + v.w;
    ss += v.x * v.x + v.y * v.y + v.z * v.z + v.w * v.w;
  }
  s = warp_allreduce_sum(s);
  ss = warp_allreduce_sum(ss);
  if (lane == 0) {
    float mu = s * (1.f / E_DIM);
    float var = ss * (1.f / E_DIM) - mu * mu;
    stats[row] = make_float2(mu, rsqrtf(var + LN_EPS));
  }
}

// ---------------------------------------------------------------------------
// Kernel 2: gx = layernorm(x) @ W_ih^T + b_ih  (M=32768, N=1536, K=512)
// Block = 256 thr (8 waves). Tile BM=64 x BN=128, BK=32, f32 WMMA 16x16x4.
// Double-buffered global->LDS staging; per-chunk fragment preload so the
// ds_load batch completes while earlier WMMAs drain.
// ---------------------------------------------------------------------------
#define BK 32
#define NCHUNK (E_DIM / BK)
#define LDSA 36  // padded row stride (floats): 16B-aligned rows, bank-clean
#define LDSB 36

__global__ __launch_bounds__(256) void gemm_gx_kernel(
    const float* __restrict__ x, const float2* __restrict__ stats,
    const float* __restrict__ Wih, const float* __restrict__ bih,
    float* __restrict__ gx) {
  __shared__ float Asm[64 * LDSA];
  __shared__ float Bsm[128 * LDSB];

  const int tid = threadIdx.x;
  const int lane = tid & 31;
  const int wid = tid >> 5;
  const int wm = wid & 3;   // 0..3 : 16-row slice
  const int wn = wid >> 2;  // 0..1 : 64-col half
  const int m0 = blockIdx.y * 64;
  const int n0 = blockIdx.x * 128;

  // cooperative-load coordinates
  const int ar = tid >> 2;          // 0..63   (A row)
  const int ac = (tid & 3) * 8;     // col segment of 8
  const int br = tid >> 1;          // 0..127  (B row)
  const int bc = (tid & 1) * 16;    // col segment of 16
  const float2 st = stats[m0 + ar]; // layernorm stats for my A row
  const float* xrow = x + (size_t)(m0 + ar) * E_DIM + ac;
  const float* wrow = Wih + (size_t)(n0 + br) * E_DIM + bc;
  float* Asw = &Asm[ar * LDSA + ac];
  float* Bsw = &Bsm[br * LDSB + bc];

  // WMMA fragment coordinates
  const int mn = lane & 15;
  const int ko = (lane >> 4) * 2;  // K-pair offset within 4-wide step
  const float* Ap = &Asm[(wm * 16 + mn) * LDSA + ko];
  const float* Bp = &Bsm[(wn * 64 + mn) * LDSB + ko];

  v8f acc[4] = {};

  // prologue: fetch chunk 0
  float4 av[2], bv[4];
#pragma unroll
  for (int q = 0; q < 2; ++q) av[q] = *(const float4*)(xrow + q * 4);
#pragma unroll
  for (int q = 0; q < 4; ++q) bv[q] = *(const float4*)(wrow + q * 4);

  for (int kc = 0; kc < NCHUNK; ++kc) {
    // stage current chunk into LDS (A normalized on the fly)
#pragma unroll
    for (int q = 0; q < 2; ++q) {
      float4 v = av[q];
      v.x = (v.x - st.x) * st.y;
      v.y = (v.y - st.x) * st.y;
      v.z = (v.z - st.x) * st.y;
      v.w = (v.w - st.x) * st.y;
      *(float4*)(Asw + q * 4) = v;
    }
#pragma unroll
    for (int q = 0; q < 4; ++q) *(float4*)(Bsw + q * 4) = bv[q];
    __syncthreads();

    // prefetch next chunk from global while WMMAs run below
    if (kc + 1 < NCHUNK) {
      const int k0 = (kc + 1) * BK;
#pragma unroll
      for (int q = 0; q < 2; ++q) av[q] = *(const float4*)(xrow + k0 + q * 4);
#pragma unroll
      for (int q = 0; q < 4; ++q) bv[q] = *(const float4*)(wrow + k0 + q * 4);
    }

    // preload all fragments of the chunk, then issue 32 WMMAs
    v2f afrag[8];
    v2f bfrag[4][8];
#pragma unroll
    for (int s = 0; s < 8; ++s) afrag[s] = *(const v2f*)(Ap + 4 * s);
#pragma unroll
    for (int j = 0; j < 4; ++j)
#pragma unroll
      for (int s = 0; s < 8; ++s)
        bfrag[j][s] = *(const v2f*)(Bp + j * 16 * LDSB + 4 * s);
#pragma unroll
    for (int s = 0; s < 8; ++s)
#pragma unroll
      for (int j = 0; j < 4; ++j)
        acc[j] = wmma_f32(afrag[s], bfrag[j][s], acc[j]);
    __syncthreads();
  }

  // epilogue: + b_ih, store (C/D layout: M = i + 8*(lane>=16), N = lane&15)
  const int koh = (lane >> 4) * 8;
#pragma unroll
  for (int j = 0; j < 4; ++j) {
    const int gcol = n0 + wn * 64 + j * 16 + mn;
    const float bias = bih[gcol];
#pragma unroll
    for (int i = 0; i < 8; ++i) {
      const int grow = m0 + wm * 16 + i + koh;
      gx[(size_t)grow * G_DIM + gcol] = acc[j][i] + bias;
    }
  }
}

// ---------------------------------------------------------------------------
// Kernel 3: one GRU time step. Each wave computes one 16(b) x 16(h) tile for
// all three gates (r,z,n) sharing A fragments. 8 waves/block, 32 blocks.
// Fragments are preloaded per 32-wide K chunk (32 global float2 loads batched
// ahead of 24 WMMAs); next chunk is prefetched via global_prefetch.
// ---------------------------------------------------------------------------
__global__ __launch_bounds__(256) void gru_step_kernel(
    const float* __restrict__ hprev, const float* __restrict__ Whh,
    const float* __restrict__ bhh, const float* __restrict__ gx_t,
    const int* __restrict__ done_t, float* __restrict__ hnext,
    float* __restrict__ ys_t) {
  const int lane = threadIdx.x & 31;
  const int wid = threadIdx.x >> 5;
  const int tile = blockIdx.x * 8 + wid;  // 0..255
  const int ht = tile & 31;
  const int bt = tile >> 5;
  const int b0 = bt * 16, h0 = ht * 16;
  const int mn = lane & 15;
  const int ko = (lane >> 4) * 2;

  v8f cr = {}, cz = {}, cn = {};
  const float* Ap = hprev + (size_t)(b0 + mn) * H_DIM + ko;
  const float* Brp = Whh + (size_t)(h0 + mn) * H_DIM + ko;  // gate r rows
  const float* Bzp = Brp + (size_t)H_DIM * H_DIM;           // gate z rows
  const float* Bnp = Bzp + (size_t)H_DIM * H_DIM;           // gate n rows

  for (int kc = 0; kc < H_DIM; kc += 32) {
    if (kc + 32 < H_DIM) {
      __builtin_prefetch(Ap + kc + 32, 0, 0);
      __builtin_prefetch(Brp + kc + 32, 0, 0);
      __builtin_prefetch(Bzp + kc + 32, 0, 0);
      __builtin_prefetch(Bnp + kc + 32, 0, 0);
    }
    v2f a[8], fr[8], fz[8], fn[8];
#pragma unroll
    for (int s = 0; s < 8; ++s) {
      a[s] = *(const v2f*)(Ap + kc + 4 * s);
      fr[s] = *(const v2f*)(Brp + kc + 4 * s);
      fz[s] = *(const v2f*)(Bzp + kc + 4 * s);
      fn[s] = *(const v2f*)(Bnp + kc + 4 * s);
    }
#pragma unroll
    for (int s = 0; s < 8; ++s) {
      cr = wmma_f32(a[s], fr[s], cr);
      cz = wmma_f32(a[s], fz[s], cz);
      cn = wmma_f32(a[s], fn[s], cn);
    }
  }

  const int col = h0 + mn;
  const float bhr = bhh[col];
  const float bhz = bhh[H_DIM + col];
  const float bhn = bhh[2 * H_DIM + col];
  const int koh = (lane >> 4) * 8;
#pragma unroll
  for (int i = 0; i < 8; ++i) {
    const int row = b0 + i + koh;
    const size_t hidx = (size_t)row * H_DIM + col;
    const size_t gidx = (size_t)row * G_DIM + col;
    const float gxr = gx_t[gidx];
    const float gxz = gx_t[gidx + H_DIM];
    const float gxn = gx_t[gidx + 2 * H_DIM];
    const float r = 1.f / (1.f + __expf(-(gxr + cr[i] + bhr)));
    const float z = 1.f / (1.f + __expf(-(gxz + cz[i] + bhz)));
    const float nv = tanhf(gxn + r * (cn[i] + bhn));
    const float hold = hprev[hidx];
    const float hnew = (1.f - z) * nv + z * hold;
    ys_t[hidx] = hnew;
    const float keep = 1.f - (float)done_t[row];
    hnext[hidx] = hnew * keep;
  }
}

// ---------------------------------------------------------------------------
// Kernel 4: in-place layernorm over last dim (H=512) of ys in d_out.
// ---------------------------------------------------------------------------
__global__ __launch_bounds__(256) void layernorm_kernel(float* __restrict__ y) {
  const int wid = threadIdx.x >> 5;
  const int lane = threadIdx.x & 31;
  const size_t row = (size_t)blockIdx.x * 8 + wid;
  float* yr = y + row * H_DIM;
  float4 v[4];
  float s = 0.f, ss = 0.f;
#pragma unroll
  for (int j = 0; j < 4; ++j) {
    v[j] = *(const float4*)(yr + lane * 4 + j * 128);
    s += v[j].x + v[j].y + v[j].z + v[j].w;
    ss += v[j].x * v[j].x + v[j].y * v[j].y + v[j].z * v[j].z + v[j].w * v[j].w;
  }
  s = warp_allreduce_sum(s);
  ss = warp_allreduce_sum(ss);
  const float mu = s * (1.f / H_DIM);
  const float rstd = rsqrtf(ss * (1.f / H_DIM) - mu * mu + LN_EPS);
#pragma unroll
  for (int j = 0; j < 4; ++j) {
    float4 o;
    o.x = (v[j].x - mu) * rstd;
    o.y = (v[j].y - mu) * rstd;
    o.z = (v[j].z - mu) * rstd;
    o.w = (v[j].w - mu) * rstd;
    *(float4*)(yr + lane * 4 + j * 128) = o;
  }
}

// ---------------------------------------------------------------------------
extern "C" void kernel_launch(void* const* d_in, const int* in_sizes, int n_in,
                              void* d_out, int out_size, void* d_ws,
                              size_t ws_size, hipStream_t stream) {
  (void)in_sizes; (void)n_in; (void)out_size; (void)ws_size;
  const float* x = (const float*)d_in[0];
  const float* hidden = (const float*)d_in[1];
  const int* done = (const int*)d_in[2];
  const float* Wih = (const float*)d_in[3];
  const float* Whh = (const float*)d_in[4];
  const float* bih = (const float*)d_in[5];
  const float* bhh = (const float*)d_in[6];
  float* out = (float*)d_out;

  const size_t NROW = (size_t)T_DIM * B_DIM;  // 32768
  char* ws = (char*)d_ws;
  float2* stats = (float2*)ws;                               // 256 KB
  float* gx = (float*)(ws + ((NROW * sizeof(float2) + 255) & ~(size_t)255));
  float* hbuf0 = gx + NROW * G_DIM;                          // after 192 MB
  float* hbuf1 = hbuf0 + (size_t)B_DIM * H_DIM;

  // 1) layernorm stats of x rows
  rowstats_kernel<<<NROW / 8, 256, 0, stream>>>(x, stats);

  // 2) gx = norm(x) @ W_ih^T + b_ih
  dim3 g2(G_DIM / 128, NROW / 64);
  gemm_gx_kernel<<<g2, 256, 0, stream>>>(x, stats, Wih, bih, gx);

  // 3) recurrent scan (ping-pong hidden state)
  hipMemcpyAsync(hbuf0, hidden, (size_t)B_DIM * H_DIM * sizeof(float),
                 hipMemcpyDeviceToDevice, stream);
  for (int t = 0; t < T_DIM; ++t) {
    const float* hp = (t & 1) ? hbuf1 : hbuf0;
    float* hn = (t & 1) ? hbuf0 : hbuf1;
    gru_step_kernel<<<32, 256, 0, stream>>>(
        hp, Whh, bhh, gx + (size_t)t * B_DIM * G_DIM, done + t * B_DIM, hn,
        out + (size_t)t * B_DIM * H_DIM);
  }
  // h_last = h after step 255 (written to hbuf0 since 255 is odd)
  hipMemcpyAsync(out + NROW * H_DIM, hbuf0,
                 (size_t)B_DIM * H_DIM * sizeof(float),
                 hipMemcpyDeviceToDevice, stream);

  // 4) y = layernorm(ys) in place
  layernorm_kernel<<<NROW / 8, 256, 0, stream>>>(out);
}